// RadialPart_21629455302717
// MI455X (gfx1250) — compile-verified
//
#include <hip/hip_runtime.h>
#include <hip/hip_bf16.h>
#include <stdint.h>

#define NB    10            // DEGREE+1
#define NU    8
#define NZ    10
#define NPAIR (NZ*NZ)       // 100
#define CSZ   (NPAIR*NU*NB) // 8000 floats = 32 KB
#define BLK   256

// CDNA5 async global->LDS staging, guarded so compile never fails.
#if defined(__gfx1250__) && __has_builtin(__builtin_amdgcn_global_load_async_to_lds_b128) && __has_builtin(__builtin_amdgcn_s_wait_asynccnt)
#define USE_ASYNC_LDS 1
#else
#define USE_ASYNC_LDS 0
#endif

typedef int v4i __attribute__((ext_vector_type(4)));

__global__ __launch_bounds__(BLK)
void radial_mtp_kernel(const float* __restrict__ r,
                       const int*   __restrict__ zi,
                       const int*   __restrict__ zj,
                       const float* __restrict__ c,
                       float*       __restrict__ out,
                       int E)
{
    // raw: c in original [pair][u][b] layout (async-DMA friendly, byte-identical copy)
    // tr : transposed [pair][b][u] so inner loop does aligned b128 LDS reads
    __shared__ float raw[CSZ];
    __shared__ float tr[CSZ];

    const int tid = threadIdx.x;

    // ---- Stage 32KB coefficient table into LDS ----
    {
        const float4* c4 = (const float4*)c;   // 2000 x 16B, device alloc is 16B aligned
        float4* raw4 = (float4*)raw;
#if USE_ASYNC_LDS
        // Builtin signature (from clang diagnostic): param0 = AS1 int4*, so pass
        // AS1-qualified v4i* for the global source and AS3-qualified v4i* for the
        // LDS destination (each implicitly converts to generic if needed).
        typedef __attribute__((address_space(1))) v4i gv4i;
        typedef __attribute__((address_space(3))) v4i lv4i;
        for (int t = tid; t < CSZ / 4; t += BLK) {
            __builtin_amdgcn_global_load_async_to_lds_b128(
                (gv4i*)(c4 + t), (lv4i*)(raw4 + t), 0, 0);
        }
        __builtin_amdgcn_s_wait_asynccnt(0);   // wave's own async ops complete
#else
        for (int t = tid; t < CSZ / 4; t += BLK) {
            raw4[t] = c4[t];
        }
#endif
    }
    __syncthreads();   // all waves' staged data visible

    // ---- Transpose [pair][u][b] -> [pair][b][u] (one-time, 8000 elems) ----
    for (int t = tid; t < CSZ; t += BLK) {
        int b = t % NB;
        int u = (t / NB) % NU;
        int p = t / (NB * NU);
        tr[(p * NB + b) * NU + u] = raw[t];
    }
    __syncthreads();

    // ---- Streaming main loop: one edge per thread per iteration ----
    const int gid    = blockIdx.x * BLK + tid;
    const int stride = gridDim.x * BLK;

    for (int e = gid; e < E; e += stride) {
        // CDNA5 streaming prefetch for next grid-stride iteration
        __builtin_prefetch(r  + e + stride, 0, 1);
        __builtin_prefetch(zi + e + stride, 0, 1);
        __builtin_prefetch(zj + e + stride, 0, 1);

        const float rv = r[e];
        const int   p  = zi[e] * NZ + zj[e];

        const float nr = rv * 0.2f;          // (r - 0) / (5 - 0)
        // Chebyshev T_0..T_9 (computed everywhere; env*mask zeroes r >= r_cut,
        // matching the reference's static-mask semantics)
        float T[NB];
        T[0] = 1.0f;
        T[1] = nr;
        const float two_nr = nr + nr;
        #pragma unroll
        for (int b = 2; b < NB; ++b)
            T[b] = fmaf(two_nr, T[b - 1], -T[b - 2]);

        float om  = 1.0f - nr;
        float env = om * om;
        env = (rv < 5.0f) ? env : 0.0f;      // mask folded into envelope

        const float* cp = tr + p * (NB * NU);   // 16B aligned (32B stride per b)

        float a0 = 0.f, a1 = 0.f, a2 = 0.f, a3 = 0.f;
        float a4 = 0.f, a5 = 0.f, a6 = 0.f, a7 = 0.f;
        #pragma unroll
        for (int b = 0; b < NB; ++b) {
            const float4 lo = *(const float4*)(cp + b * NU);
            const float4 hi = *(const float4*)(cp + b * NU + 4);
            const float t  = T[b];
            a0 = fmaf(lo.x, t, a0);  a1 = fmaf(lo.y, t, a1);
            a2 = fmaf(lo.z, t, a2);  a3 = fmaf(lo.w, t, a3);
            a4 = fmaf(hi.x, t, a4);  a5 = fmaf(hi.y, t, a5);
            a6 = fmaf(hi.z, t, a6);  a7 = fmaf(hi.w, t, a7);
        }

        float4 o0 = make_float4(a0 * env, a1 * env, a2 * env, a3 * env);
        float4 o1 = make_float4(a4 * env, a5 * env, a6 * env, a7 * env);
        float4* op = (float4*)(out + (size_t)e * NU);   // 32B aligned, coalesced
        op[0] = o0;
        op[1] = o1;
    }
}

extern "C" void kernel_launch(void* const* d_in, const int* in_sizes, int n_in,
                              void* d_out, int out_size, void* d_ws, size_t ws_size,
                              hipStream_t stream) {
    const float* r  = (const float*)d_in[0];
    const int*   zi = (const int*)d_in[1];
    const int*   zj = (const int*)d_in[2];
    const float* c  = (const float*)d_in[3];
    float* out = (float*)d_out;
    const int E = in_sizes[0];

    int blocks = (E + BLK - 1) / BLK;
    if (blocks > 1536) blocks = 1536;   // ~8+ edges/thread to amortize LDS staging

    hipLaunchKernelGGL(radial_mtp_kernel, dim3(blocks), dim3(BLK), 0, stream,
                       r, zi, zj, c, out, E);
}